// GraphSAGE_64355789963304
// MI455X (gfx1250) — compile-verified
//
#include <hip/hip_runtime.h>
#include <cmath>

typedef __attribute__((ext_vector_type(2))) float v2f;
typedef __attribute__((ext_vector_type(8))) float v8f;

#define NNODES 50000
#define NEDGES 800000

__device__ __forceinline__ float elu1(float x) {
  return x > 0.0f ? x : expm1f(x);
}

// ---------------- zero fill (vectorized, n must be multiple of 4) -------
__global__ void zero_f32v4(float4* __restrict__ p, long n4) {
  long i = (long)blockIdx.x * blockDim.x + threadIdx.x;
  long stride = (long)gridDim.x * blockDim.x;
  const float4 z = {0.0f, 0.0f, 0.0f, 0.0f};
  for (; i < n4; i += stride) p[i] = z;
}

// ---------------- degree count ----------------
__global__ void degree_kernel(const long long* __restrict__ ei,
                              float* __restrict__ deg) {
  int e = blockIdx.x * blockDim.x + threadIdx.x;
  if (e < NEDGES) {
    int d = (int)ei[NEDGES + e];
    unsafeAtomicAdd(&deg[d], 1.0f);
  }
}

__global__ void deginv_kernel(float* __restrict__ deg) {
  int i = blockIdx.x * blockDim.x + threadIdx.x;
  if (i < NNODES) deg[i] = 1.0f / fmaxf(deg[i], 1.0f);
}

// ---------------- edge scatter-add ----------------
// One wave (32 lanes) per edge. Edge id is wave-uniform -> readfirstlane so
// the index loads become scalar (s_load_b64). Each of the 4 atomic
// instructions covers a contiguous 128B segment (lane*4 + i*128).
__global__ __launch_bounds__(256) void scatter_kernel(
    const float* __restrict__ feat, const long long* __restrict__ ei,
    float* __restrict__ agg) {
  int wave_in_blk = threadIdx.x >> 5;
  int lane = threadIdx.x & 31;
  int e = __builtin_amdgcn_readfirstlane(blockIdx.x * 8 + wave_in_blk);
  if (e >= NEDGES) return;
  int s = (int)ei[e];
  int d = (int)ei[NEDGES + e];
  const float* srcp = feat + (long)s * 128 + lane;
  float* dstp = agg + (long)d * 128 + lane;
  float v0 = srcp[0];
  float v1 = srcp[32];
  float v2 = srcp[64];
  float v3 = srcp[96];
  unsafeAtomicAdd(dstp + 0,  v0);
  unsafeAtomicAdd(dstp + 32, v1);
  unsafeAtomicAdd(dstp + 64, v2);
  unsafeAtomicAdd(dstp + 96, v3);
}

// ---------------- layer 1: h1 = elu([agg/deg|x] @ [Wl;Wr] + b) * mask ----
// block = 128 threads (4 waves), each wave owns 2 column tiles of 16.
__global__ __launch_bounds__(128) void sage_gemm128(
    const float* __restrict__ agg, const float* __restrict__ deginv,
    const float* __restrict__ xin, const float* __restrict__ Wl,
    const float* __restrict__ bias, const float* __restrict__ Wr,
    const float* __restrict__ mask, float* __restrict__ out) {
  __shared__ float As[16 * 256];  // 16 rows x (128 agg-scaled | 128 x)
  const int tid = threadIdx.x;
  const long row0 = (long)blockIdx.x * 16;

  for (int i = tid; i < 16 * 64; i += 128) {
    int r = i >> 6;
    int c4 = (i & 63) << 2;
    long gr = row0 + r;
    float4 v;
    if (c4 < 128) {
      v = *(const float4*)(agg + gr * 128 + c4);
      float s = deginv[gr];
      v.x *= s; v.y *= s; v.z *= s; v.w *= s;
    } else {
      v = *(const float4*)(xin + gr * 128 + (c4 - 128));
    }
    *(float4*)(&As[r * 256 + c4]) = v;
  }
  __syncthreads();

  const int wave = tid >> 5, lane = tid & 31;
  const int half = lane >> 4, ln = lane & 15;
  const int n0 = wave * 32 + ln, n1 = n0 + 16;
  const float* arow = As + ln * 256;

  v8f acc0 = {}, acc1 = {};
  // K = 0..127 : A = scaled aggregation, B = Wl
  for (int kk = 0; kk < 32; ++kk) {
    int k = (kk << 2) + (half << 1);
    v2f a = { arow[k], arow[k + 1] };
    v2f b0 = { Wl[k * 128 + n0], Wl[(k + 1) * 128 + n0] };
    v2f b1 = { Wl[k * 128 + n1], Wl[(k + 1) * 128 + n1] };
    acc0 = __builtin_amdgcn_wmma_f32_16x16x4_f32(false, a, false, b0, (short)0, acc0, false, false);
    acc1 = __builtin_amdgcn_wmma_f32_16x16x4_f32(false, a, false, b1, (short)0, acc1, false, false);
  }
  // K = 128..255 : A = x, B = Wr
  for (int kk = 0; kk < 32; ++kk) {
    int k = (kk << 2) + (half << 1);
    v2f a = { arow[128 + k], arow[128 + k + 1] };
    v2f b0 = { Wr[k * 128 + n0], Wr[(k + 1) * 128 + n0] };
    v2f b1 = { Wr[k * 128 + n1], Wr[(k + 1) * 128 + n1] };
    acc0 = __builtin_amdgcn_wmma_f32_16x16x4_f32(false, a, false, b0, (short)0, acc0, false, false);
    acc1 = __builtin_amdgcn_wmma_f32_16x16x4_f32(false, a, false, b1, (short)0, acc1, false, false);
  }

  const float bv0 = bias[n0], bv1 = bias[n1];
  for (int r = 0; r < 8; ++r) {
    long m = row0 + r + (half << 3);
    float v0 = elu1(acc0[r] + bv0) * mask[m * 128 + n0];
    float v1 = elu1(acc1[r] + bv1) * mask[m * 128 + n1];
    out[m * 128 + n0] = v0;
    out[m * 128 + n1] = v1;
  }
}

// ---------------- layer 2 + log_softmax (out dim 32, 1 wave/block) ------
__global__ __launch_bounds__(32) void sage_gemm32_lsm(
    const float* __restrict__ agg, const float* __restrict__ deginv,
    const float* __restrict__ hin, const float* __restrict__ Wl,
    const float* __restrict__ bias, const float* __restrict__ Wr,
    const float* __restrict__ mask, float* __restrict__ out) {
  __shared__ float As[16 * 256];
  const int lane = threadIdx.x;
  const long row0 = (long)blockIdx.x * 16;

  for (int i = lane; i < 16 * 64; i += 32) {
    int r = i >> 6;
    int c4 = (i & 63) << 2;
    long gr = row0 + r;
    float4 v;
    if (c4 < 128) {
      v = *(const float4*)(agg + gr * 128 + c4);
      float s = deginv[gr];
      v.x *= s; v.y *= s; v.z *= s; v.w *= s;
    } else {
      v = *(const float4*)(hin + gr * 128 + (c4 - 128));
    }
    *(float4*)(&As[r * 256 + c4]) = v;
  }
  __syncthreads();

  const int half = lane >> 4, ln = lane & 15;
  const float* arow = As + ln * 256;

  v8f acc0 = {}, acc1 = {};
  for (int kk = 0; kk < 32; ++kk) {
    int k = (kk << 2) + (half << 1);
    v2f a = { arow[k], arow[k + 1] };
    v2f b0 = { Wl[k * 32 + ln], Wl[(k + 1) * 32 + ln] };
    v2f b1 = { Wl[k * 32 + 16 + ln], Wl[(k + 1) * 32 + 16 + ln] };
    acc0 = __builtin_amdgcn_wmma_f32_16x16x4_f32(false, a, false, b0, (short)0, acc0, false, false);
    acc1 = __builtin_amdgcn_wmma_f32_16x16x4_f32(false, a, false, b1, (short)0, acc1, false, false);
  }
  for (int kk = 0; kk < 32; ++kk) {
    int k = (kk << 2) + (half << 1);
    v2f a = { arow[128 + k], arow[128 + k + 1] };
    v2f b0 = { Wr[k * 32 + ln], Wr[(k + 1) * 32 + ln] };
    v2f b1 = { Wr[k * 32 + 16 + ln], Wr[(k + 1) * 32 + 16 + ln] };
    acc0 = __builtin_amdgcn_wmma_f32_16x16x4_f32(false, a, false, b0, (short)0, acc0, false, false);
    acc1 = __builtin_amdgcn_wmma_f32_16x16x4_f32(false, a, false, b1, (short)0, acc1, false, false);
  }

  const float bv0 = bias[ln], bv1 = bias[16 + ln];
  for (int r = 0; r < 8; ++r) {
    long m = row0 + r + (half << 3);
    float v0 = elu1(acc0[r] + bv0) * mask[m * 32 + ln];
    float v1 = elu1(acc1[r] + bv1) * mask[m * 32 + 16 + ln];
    // log-softmax over the 32-col row: values live in 16 lanes x 2 regs,
    // xor masks 1..8 stay inside each 16-lane half (wave32).
    float mx = fmaxf(v0, v1);
    for (int off = 8; off > 0; off >>= 1) mx = fmaxf(mx, __shfl_xor(mx, off, 32));
    float s = expf(v0 - mx) + expf(v1 - mx);
    for (int off = 8; off > 0; off >>= 1) s += __shfl_xor(s, off, 32);
    float lse = mx + logf(s);
    out[m * 32 + ln] = v0 - lse;
    out[m * 32 + 16 + ln] = v1 - lse;
  }
}

extern "C" void kernel_launch(void* const* d_in, const int* in_sizes, int n_in,
                              void* d_out, int out_size, void* d_ws, size_t ws_size,
                              hipStream_t stream) {
  const float*     x   = (const float*)d_in[0];
  const long long* ei  = (const long long*)d_in[1];   // [2, E] int64
  const float*     W1l = (const float*)d_in[2];
  const float*     b1  = (const float*)d_in[3];
  const float*     W1r = (const float*)d_in[4];
  const float*     W2l = (const float*)d_in[5];
  const float*     b2  = (const float*)d_in[6];
  const float*     W2r = (const float*)d_in[7];
  const float*     m1  = (const float*)d_in[8];
  const float*     m2  = (const float*)d_in[9];
  float* out = (float*)d_out;

  // workspace layout (floats): agg[N*128] | deg[N] | h1[N*128]  (~51.4 MB)
  float* agg = (float*)d_ws;
  float* deg = agg + (long)NNODES * 128;
  float* h1  = deg + NNODES;

  const int ZB = 2048, TB = 256;
  // one wave per edge, 8 waves (256 threads) per block
  int scat_blocks = (NEDGES + 7) / 8;

  // layer 1
  zero_f32v4<<<ZB, TB, 0, stream>>>((float4*)agg,
                                    ((long)NNODES * 128 + NNODES) / 4);
  degree_kernel<<<(NEDGES + TB - 1) / TB, TB, 0, stream>>>(ei, deg);
  deginv_kernel<<<(NNODES + TB - 1) / TB, TB, 0, stream>>>(deg);
  scatter_kernel<<<scat_blocks, TB, 0, stream>>>(x, ei, agg);
  sage_gemm128<<<NNODES / 16, 128, 0, stream>>>(agg, deg, x, W1l, b1, W1r, m1, h1);

  // layer 2 (reuse agg)
  zero_f32v4<<<ZB, TB, 0, stream>>>((float4*)agg, (long)NNODES * 128 / 4);
  scatter_kernel<<<scat_blocks, TB, 0, stream>>>(h1, ei, agg);
  sage_gemm32_lsm<<<NNODES / 16, 32, 0, stream>>>(agg, deg, h1, W2l, b2, W2r, m2, out);
}